// DiffPooling_78821239816695
// MI455X (gfx1250) — compile-verified
//
#include <hip/hip_runtime.h>
#include <math.h>

#define N_NODES 8192
#define D_FEAT  512
#define K_CLUS  256

typedef __attribute__((ext_vector_type(2))) float v2f;
typedef __attribute__((ext_vector_type(8))) float v8f;

// D = A(16x4) * B(4x16) + C  -- CDNA5 f32 WMMA
__device__ __forceinline__ v8f wmma4(v2f a, v2f b, v8f c) {
  return __builtin_amdgcn_wmma_f32_16x16x4_f32(false, a, false, b, (short)0, c,
                                               false, false);
}

// 256-thread block sum; result valid in thread 0.
__device__ __forceinline__ float blockSum256(float v, float* sh8) {
#pragma unroll
  for (int o = 16; o; o >>= 1) v += __shfl_xor(v, o, 32);
  int w = threadIdx.x >> 5, l = threadIdx.x & 31;
  if (l == 0) sh8[w] = v;
  __syncthreads();
  v = (threadIdx.x < 8) ? sh8[threadIdx.x] : 0.f;
  if (w == 0) {
#pragma unroll
    for (int o = 4; o; o >>= 1) v += __shfl_xor(v, o, 32);
  }
  __syncthreads();
  return v;
}

__global__ void k_zero(float* __restrict__ p, int n) {
  int i = blockIdx.x * blockDim.x + threadIdx.x;
  int stride = gridDim.x * blockDim.x;
  for (; i < n; i += stride) p[i] = 0.f;
}

// logits = F@W + b, row-wise softmax -> assignments; per-column sums -> g_cs.
// 32 rows x 256 cols per block, 8 waves.
__global__ __launch_bounds__(256) void k_logits_softmax(
    const float* __restrict__ F, const float* __restrict__ W,
    const float* __restrict__ bias, float* __restrict__ assign,
    float* __restrict__ g_cs) {
  __shared__ float As[32][16];
  __shared__ float Bs[16][256];
  __shared__ float Ls[32][257];
  __shared__ float colsum[256];
  const int tid = threadIdx.x;
  const int w = tid >> 5, lane = tid & 31;
  const int half = lane >> 4, mrow = lane & 15, kb = half * 2;
  const int row0 = blockIdx.x * 32;
  colsum[tid] = 0.f;

  const int rt = w >> 2;        // row tile 0..1
  const int ct0 = (w & 3) * 4;  // col tiles ct0..ct0+3
  const v8f vz = {0, 0, 0, 0, 0, 0, 0, 0};
  v8f acc[4];
#pragma unroll
  for (int t = 0; t < 4; ++t) acc[t] = vz;

  for (int k0 = 0; k0 < D_FEAT; k0 += 16) {
    {
      int idx = tid;  // 32x16 = 512 elems
#pragma unroll
      for (int q = 0; q < 2; ++q) {
        int r = idx >> 4, c = idx & 15;
        As[r][c] = F[(size_t)(row0 + r) * D_FEAT + k0 + c];
        idx += 256;
      }
    }
    {
      int idx = tid;  // 16x256 = 4096 elems
#pragma unroll
      for (int q = 0; q < 16; ++q) {
        int r = idx >> 8, c = idx & 255;
        Bs[r][c] = W[(size_t)(k0 + r) * K_CLUS + c];
        idx += 256;
      }
    }
    __syncthreads();
#pragma unroll
    for (int kk = 0; kk < 16; kk += 4) {
      v2f a;
      a.x = As[rt * 16 + mrow][kk + kb];
      a.y = As[rt * 16 + mrow][kk + kb + 1];
#pragma unroll
      for (int t = 0; t < 4; ++t) {
        int c = (ct0 + t) * 16 + mrow;
        v2f b;
        b.x = Bs[kk + kb][c];
        b.y = Bs[kk + kb + 1][c];
        acc[t] = wmma4(a, b, acc[t]);
      }
    }
    __syncthreads();
  }
#pragma unroll
  for (int t = 0; t < 4; ++t) {
    int c = (ct0 + t) * 16 + mrow;
#pragma unroll
    for (int v = 0; v < 8; ++v)
      Ls[rt * 16 + v + 8 * half][c] = acc[t][v] + bias[c];
  }
  __syncthreads();
  // softmax: wave w owns rows w*4 .. w*4+3; 8 cols per lane.
#pragma unroll
  for (int rr = 0; rr < 4; ++rr) {
    int r = w * 4 + rr;
    float v[8];
    float m = -3.4e38f;
#pragma unroll
    for (int q = 0; q < 8; ++q) {
      v[q] = Ls[r][lane + 32 * q];
      m = fmaxf(m, v[q]);
    }
#pragma unroll
    for (int o = 16; o; o >>= 1) m = fmaxf(m, __shfl_xor(m, o, 32));
    float s = 0.f;
#pragma unroll
    for (int q = 0; q < 8; ++q) {
      v[q] = __expf(v[q] - m);
      s += v[q];
    }
#pragma unroll
    for (int o = 16; o; o >>= 1) s += __shfl_xor(s, o, 32);
    float inv = 1.f / s;
#pragma unroll
    for (int q = 0; q < 8; ++q) {
      float e = v[q] * inv;
      int c = lane + 32 * q;
      assign[(size_t)(row0 + r) * K_CLUS + c] = e;
      atomicAdd(&colsum[c], e);
    }
  }
  __syncthreads();
  atomicAdd(&g_cs[tid], colsum[tid]);
}

// sp = assign / (cs + 1e-8); entropy partial sum -> scal[1]
__global__ __launch_bounds__(256) void k_sp_ent(
    const float* __restrict__ assign, const float* __restrict__ g_cs,
    float* __restrict__ sp, float* __restrict__ scal) {
  __shared__ float sh8[8];
  float local = 0.f;
  size_t stride = (size_t)gridDim.x * blockDim.x;
  for (size_t i = (size_t)blockIdx.x * blockDim.x + threadIdx.x;
       i < (size_t)N_NODES * K_CLUS; i += stride) {
    int c = (int)(i & (K_CLUS - 1));
    float a = assign[i];
    float s = a / (g_cs[c] + 1e-8f);
    sp[i] = s;
    local += -s * __logf(s + 1e-15f);
  }
  float tot = blockSum256(local, sh8);
  if (threadIdx.x == 0) atomicAdd(&scal[1], tot);
}

// T = sp^T @ A  (256 x 8192), fused ||A||^2 -> scal[0].
// Block: full 256 k-rows x 64 cols of A. A read exactly once globally.
__global__ __launch_bounds__(256) void k_bigT(
    const float* __restrict__ sp, const float* __restrict__ A,
    float* __restrict__ T, float* __restrict__ scal) {
  __shared__ float Ss[32][257];
  __shared__ float Aa[32][65];
  __shared__ float sh8[8];
  const int tid = threadIdx.x;
  const int w = tid >> 5, lane = tid & 31;
  const int half = lane >> 4, mrow = lane & 15, kb = half * 2;
  const int j0 = blockIdx.x * 64;
  const int kt0 = w * 2;  // 2 k-tiles per wave
  const v8f vz = {0, 0, 0, 0, 0, 0, 0, 0};
  v8f acc[2][4];
#pragma unroll
  for (int t = 0; t < 2; ++t)
#pragma unroll
    for (int j = 0; j < 4; ++j) acc[t][j] = vz;
  float na2 = 0.f;
  for (int i0 = 0; i0 < N_NODES; i0 += 32) {
    {
      int idx = tid;  // 32x256 sp chunk
#pragma unroll
      for (int q = 0; q < 32; ++q) {
        int r = idx >> 8, c = idx & 255;
        Ss[r][c] = sp[(size_t)(i0 + r) * K_CLUS + c];
        idx += 256;
      }
    }
    {
      int idx = tid;  // 32x64 A chunk, fuse sum of squares
#pragma unroll
      for (int q = 0; q < 8; ++q) {
        int r = idx >> 6, c = idx & 63;
        float vv = A[(size_t)(i0 + r) * N_NODES + j0 + c];
        Aa[r][c] = vv;
        na2 += vv * vv;
        idx += 256;
      }
    }
    if (i0 + 32 < N_NODES)
      __builtin_prefetch(&A[(size_t)(i0 + 32 + (tid >> 6)) * N_NODES + j0 + (tid & 63)], 0, 1);
    __syncthreads();
#pragma unroll
    for (int kk = 0; kk < 32; kk += 4) {
#pragma unroll
      for (int t = 0; t < 2; ++t) {
        v2f a;  // A-op[m][i] = sp[i][m] (transposed LDS read)
        a.x = Ss[kk + kb][(kt0 + t) * 16 + mrow];
        a.y = Ss[kk + kb + 1][(kt0 + t) * 16 + mrow];
#pragma unroll
        for (int j = 0; j < 4; ++j) {
          v2f b;
          b.x = Aa[kk + kb][j * 16 + mrow];
          b.y = Aa[kk + kb + 1][j * 16 + mrow];
          acc[t][j] = wmma4(a, b, acc[t][j]);
        }
      }
    }
    __syncthreads();
  }
#pragma unroll
  for (int t = 0; t < 2; ++t)
#pragma unroll
    for (int j = 0; j < 4; ++j)
#pragma unroll
      for (int v = 0; v < 8; ++v) {
        int kr = (kt0 + t) * 16 + v + 8 * half;
        int jc = j0 + j * 16 + mrow;
        T[(size_t)kr * N_NODES + jc] = acc[t][j][v];
      }
  float tot = blockSum256(na2, sh8);
  if (threadIdx.x == 0) atomicAdd(&scal[0], tot);
}

// Generic: C[m,n] += sum_i L(m,i)*B[i,n], M=256, 64x64 tile/block,
// reduction split across gridDim.y with atomic accumulation.
// Lt=1: L(m,i) = L[i*ldL + m]; Lt=0: L(m,i) = L[m*ldL + i].
__global__ __launch_bounds__(256) void k_gemm_acc(
    const float* __restrict__ L, int ldL, int Lt, const float* __restrict__ B,
    int ldB, float* __restrict__ C, int ldC, int ncols, int iTotal) {
  __shared__ float Ls[32][65];
  __shared__ float Bsh[32][65];
  const int tid = threadIdx.x;
  const int w = tid >> 5, lane = tid & 31;
  const int half = lane >> 4, mrow = lane & 15, kb = half * 2;
  const int nb = ncols >> 6;
  const int m0 = (blockIdx.x / nb) * 64;
  const int n0 = (blockIdx.x % nb) * 64;
  const int ilen = iTotal / gridDim.y;
  const int ibeg = blockIdx.y * ilen;
  const int mt = w >> 1;
  const int nt0 = (w & 1) * 2;
  const v8f vz = {0, 0, 0, 0, 0, 0, 0, 0};
  v8f acc[2];
  acc[0] = vz;
  acc[1] = vz;
  for (int i0 = ibeg; i0 < ibeg + ilen; i0 += 32) {
    if (Lt) {
      int idx = tid;
#pragma unroll
      for (int q = 0; q < 8; ++q) {
        int il = idx >> 6, ml = idx & 63;
        Ls[il][ml] = L[(size_t)(i0 + il) * ldL + m0 + ml];
        idx += 256;
      }
    } else {
      int idx = tid;
#pragma unroll
      for (int q = 0; q < 8; ++q) {
        int il = idx >> 6, ml = idx & 63;
        Ls[il][ml] = L[(size_t)(m0 + ml) * ldL + i0 + il];
        idx += 256;
      }
    }
    {
      int idx = tid;
#pragma unroll
      for (int q = 0; q < 8; ++q) {
        int il = idx >> 6, nl = idx & 63;
        Bsh[il][nl] = B[(size_t)(i0 + il) * ldB + n0 + nl];
        idx += 256;
      }
    }
    __syncthreads();
#pragma unroll
    for (int kk = 0; kk < 32; kk += 4) {
      v2f a;
      a.x = Ls[kk + kb][mt * 16 + mrow];
      a.y = Ls[kk + kb + 1][mt * 16 + mrow];
#pragma unroll
      for (int j = 0; j < 2; ++j) {
        v2f b;
        b.x = Bsh[kk + kb][(nt0 + j) * 16 + mrow];
        b.y = Bsh[kk + kb + 1][(nt0 + j) * 16 + mrow];
        acc[j] = wmma4(a, b, acc[j]);
      }
    }
    __syncthreads();
  }
#pragma unroll
  for (int j = 0; j < 2; ++j)
#pragma unroll
    for (int v = 0; v < 8; ++v) {
      int mr = m0 + mt * 16 + v + 8 * half;
      int nc = n0 + (nt0 + j) * 16 + mrow;
      atomicAdd(&C[(size_t)mr * ldC + nc], acc[j][v]);
    }
}

// link = sqrt(||A||^2 - 2*trace(P) + ||G||^2)/N^2 ; ent = ent_sum / N
__global__ __launch_bounds__(256) void k_finalize(const float* __restrict__ G,
                                                  const float* __restrict__ P,
                                                  const float* __restrict__ scal,
                                                  float* __restrict__ losses) {
  __shared__ float sh8[8];
  float s = 0.f;
  for (int i = threadIdx.x; i < K_CLUS * K_CLUS; i += 256) {
    float g = G[i];
    s += g * g;
  }
  float sumG2 = blockSum256(s, sh8);
  float t = P[threadIdx.x * K_CLUS + threadIdx.x];
  float traceP = blockSum256(t, sh8);
  if (threadIdx.x == 0) {
    float ss = scal[0] - 2.f * traceP + sumG2;
    if (ss < 0.f) ss = 0.f;
    losses[0] = sqrtf(ss) / ((float)N_NODES * (float)N_NODES);
    losses[1] = scal[1] / (float)N_NODES;
  }
}

extern "C" void kernel_launch(void* const* d_in, const int* in_sizes, int n_in,
                              void* d_out, int out_size, void* d_ws,
                              size_t ws_size, hipStream_t stream) {
  const float* F = (const float*)d_in[0];
  const float* A = (const float*)d_in[1];
  const float* W = (const float*)d_in[2];
  const float* bias = (const float*)d_in[3];

  float* out = (float*)d_out;
  float* out_fp = out;                                       // 256*512
  float* out_assign = out + (size_t)K_CLUS * D_FEAT;         // 8192*256
  float* out_P = out_assign + (size_t)N_NODES * K_CLUS;      // 256*256
  float* out_loss = out_P + (size_t)K_CLUS * K_CLUS;         // 2

  float* ws = (float*)d_ws;
  float* ws_cs = ws;          // 256 cluster sizes
  float* ws_scal = ws + 256;  // [0]=||A||^2, [1]=entropy sum
  float* ws_sp = ws + 512;                                   // 8192*256
  float* ws_T = ws_sp + (size_t)N_NODES * K_CLUS;            // 256*8192
  float* ws_G = ws_T + (size_t)K_CLUS * N_NODES;             // 256*256

  k_zero<<<4, 256, 0, stream>>>(ws, 512);
  k_zero<<<64, 256, 0, stream>>>(ws_G, K_CLUS * K_CLUS);
  k_zero<<<128, 256, 0, stream>>>(out_fp, K_CLUS * D_FEAT);
  k_zero<<<64, 256, 0, stream>>>(out_P, K_CLUS * K_CLUS);

  k_logits_softmax<<<N_NODES / 32, 256, 0, stream>>>(F, W, bias, out_assign,
                                                     ws_cs);
  k_sp_ent<<<1024, 256, 0, stream>>>(out_assign, ws_cs, ws_sp, ws_scal);
  k_bigT<<<N_NODES / 64, 256, 0, stream>>>(ws_sp, A, ws_T, ws_scal);
  // features_pooled = sp^T @ F
  k_gemm_acc<<<dim3(32, 4), 256, 0, stream>>>(ws_sp, K_CLUS, 1, F, D_FEAT,
                                              out_fp, D_FEAT, D_FEAT, N_NODES);
  // adjacency_pooled = T @ sp
  k_gemm_acc<<<dim3(16, 8), 256, 0, stream>>>(ws_T, N_NODES, 0, ws_sp, K_CLUS,
                                              out_P, K_CLUS, K_CLUS, N_NODES);
  // G = sp^T @ sp
  k_gemm_acc<<<dim3(16, 8), 256, 0, stream>>>(ws_sp, K_CLUS, 1, ws_sp, K_CLUS,
                                              ws_G, K_CLUS, K_CLUS, N_NODES);
  k_finalize<<<1, 256, 0, stream>>>(ws_G, out_P, ws_scal, out_loss);
}